// PhasorLayer_3195455668476
// MI455X (gfx1250) — compile-verified
//
#include <hip/hip_runtime.h>
#include <math.h>

typedef float v2f __attribute__((ext_vector_type(2)));
typedef float v4f __attribute__((ext_vector_type(4)));
typedef float v8f __attribute__((ext_vector_type(8)));
typedef int   v4i __attribute__((ext_vector_type(4)));

#define BROWS 8192
#define KDIM  4096
#define PP    64
#define PI_F  3.14159265358979323846f
#define LN_EPS_F 1e-5f

// Async global->LDS staging (gfx1250 GLOBAL_LOAD_ASYNC_TO_LDS_B128), hedged.
#if defined(__gfx1250__) && __has_builtin(__builtin_amdgcn_global_load_async_to_lds_b128)
#define USE_ASYNC_LDS 1
typedef __attribute__((address_space(1))) v4i* gl_v4i_p;   // global AS
typedef __attribute__((address_space(3))) v4i* lds_v4i_p;  // LDS AS
#else
#define USE_ASYNC_LDS 0
#endif

static __device__ __forceinline__ void wait_async_lds() {
#if USE_ASYNC_LDS
#if __has_builtin(__builtin_amdgcn_s_wait_asynccnt)
    __builtin_amdgcn_s_wait_asynccnt(0);
#else
    asm volatile("s_wait_asynccnt 0" ::: "memory");
#endif
#endif
}

// ---------------------------------------------------------------------------
// Tiled f32-WMMA GEMM:  out[m,n] = sum_k A[m,k] * W[n,k]
//   (+ bias[n])  (* rowscale[m])  (+ residual[m,n])
// A: [BROWS, KDIM] row-major.  W: [N, KDIM] row-major.  out/residual ld = N.
// Tile: BM=256, BN=64, BK=32. 256 threads = 8 waves; wave w owns rows
// [w*32, w*32+32) x 64 cols = 2x4 grid of 16x16 f32 WMMA accumulators.
// LDS tiles stored as (k,k+1) float2 pairs so every WMMA fragment is one
// aligned ds_load_b64 straight into an even VGPR pair (no v_mov fixups).
// ---------------------------------------------------------------------------
__global__ __launch_bounds__(256)
void phasor_gemm_xWT(const float* __restrict__ A,
                     const float* __restrict__ W,
                     const float* __restrict__ bias,
                     const float* __restrict__ rowscale,
                     const float* __restrict__ residual,
                     float* __restrict__ out,
                     int N)
{
    __shared__ v2f AsP[256][17];   // AsP[m][kp] = A[m][k0+2kp .. +1], pad 17
    __shared__ v2f BsP[16][66];    // BsP[kp][n] = W[n][k0+2kp .. +1], pad 66

    const int tid  = threadIdx.x;
    const int wave = tid >> 5;          // 0..7
    const int lane = tid & 31;          // wave32
    const int half = lane >> 4;         // 0: lanes 0-15, 1: lanes 16-31
    const int lr   = lane & 15;
    const int m_base = wave * 32;

    const int n0 = blockIdx.x * 64;
    const int m0 = blockIdx.y * 256;

    v8f acc[2][4];
    #pragma unroll
    for (int mt = 0; mt < 2; ++mt)
        #pragma unroll
        for (int nt = 0; nt < 4; ++nt) acc[mt][nt] = 0.0f;

    for (int k0 = 0; k0 < KDIM; k0 += 32) {
        // ---- stage A tile: 256 rows x 32 cols = 2048 x 16B, 8 per thread
        #pragma unroll
        for (int i = 0; i < 8; ++i) {
            int q   = tid + i * 256;
            int row = q >> 3;
            int c4  = q & 7;
            const float* gsrc = A + (size_t)(m0 + row) * KDIM + k0 + c4 * 4;
#if USE_ASYNC_LDS
            __builtin_amdgcn_global_load_async_to_lds_b128(
                (gl_v4i_p)const_cast<float*>(gsrc),
                (lds_v4i_p)&AsP[row][c4 * 2], 0, 0);
#else
            v4f av = *reinterpret_cast<const v4f*>(gsrc);
            AsP[row][c4 * 2 + 0] = v2f{av.x, av.y};
            AsP[row][c4 * 2 + 1] = v2f{av.z, av.w};
#endif
        }
        // ---- stage B tile transposed into k-pairs: 64 W-rows x 32 cols
        #pragma unroll
        for (int i = 0; i < 2; ++i) {
            int q  = tid + i * 256;
            int nr = q >> 3;
            int c4 = q & 7;
            v4f wv = *reinterpret_cast<const v4f*>(
                W + (size_t)(n0 + nr) * KDIM + k0 + c4 * 4);
            BsP[c4 * 2 + 0][nr] = v2f{wv.x, wv.y};
            BsP[c4 * 2 + 1][nr] = v2f{wv.z, wv.w};
        }
        // ---- prefetch next K-slice (global_prefetch_b8)
        if (k0 + 32 < KDIM) {
            __builtin_prefetch(A + (size_t)(m0 + tid) * KDIM + k0 + 32, 0, 1);
            __builtin_prefetch(W + (size_t)(n0 + (tid >> 2)) * KDIM
                                 + k0 + 32 + (tid & 3) * 8, 0, 1);
        }
        wait_async_lds();
        __syncthreads();

        // ---- 8 WMMA K-steps of 4 (kp pairs step 2); 8 WMMAs per step
        #pragma unroll
        for (int kpi = 0; kpi < 16; kpi += 2) {
            // A frags (16x4 f32): lane half selects pair kpi / kpi+1
            v2f a0 = AsP[m_base +      lr][kpi + half];
            v2f a1 = AsP[m_base + 16 + lr][kpi + half];
            v2f b0 = BsP[kpi + half][ 0 + lr];
            v2f b1 = BsP[kpi + half][16 + lr];
            v2f b2 = BsP[kpi + half][32 + lr];
            v2f b3 = BsP[kpi + half][48 + lr];
            acc[0][0] = __builtin_amdgcn_wmma_f32_16x16x4_f32(false, a0, false, b0, (short)0, acc[0][0], false, false);
            acc[0][1] = __builtin_amdgcn_wmma_f32_16x16x4_f32(false, a0, false, b1, (short)0, acc[0][1], false, false);
            acc[0][2] = __builtin_amdgcn_wmma_f32_16x16x4_f32(false, a0, false, b2, (short)0, acc[0][2], false, false);
            acc[0][3] = __builtin_amdgcn_wmma_f32_16x16x4_f32(false, a0, false, b3, (short)0, acc[0][3], false, false);
            acc[1][0] = __builtin_amdgcn_wmma_f32_16x16x4_f32(false, a1, false, b0, (short)0, acc[1][0], false, false);
            acc[1][1] = __builtin_amdgcn_wmma_f32_16x16x4_f32(false, a1, false, b1, (short)0, acc[1][1], false, false);
            acc[1][2] = __builtin_amdgcn_wmma_f32_16x16x4_f32(false, a1, false, b2, (short)0, acc[1][2], false, false);
            acc[1][3] = __builtin_amdgcn_wmma_f32_16x16x4_f32(false, a1, false, b3, (short)0, acc[1][3], false, false);
        }
        __syncthreads();
    }

    // ---- epilogue; C/D layout: VGPR r -> M = r + 8*half, lanes carry N
    #pragma unroll
    for (int nt = 0; nt < 4; ++nt) {
        int col = n0 + nt * 16 + lr;
        float bval = bias ? bias[col] : 0.0f;
        #pragma unroll
        for (int mt = 0; mt < 2; ++mt) {
            #pragma unroll
            for (int r = 0; r < 8; ++r) {
                int row = m0 + m_base + mt * 16 + half * 8 + r;
                float v = acc[mt][nt][r] + bval;
                if (rowscale) v *= rowscale[row];
                if (residual) v += residual[(size_t)row * N + col];
                out[(size_t)row * N + col] = v;
            }
        }
    }
}

// ---------------------------------------------------------------------------
// Per-row phase alignment: one wave (32 lanes) per row, 8 rows per block.
// scale[b] = align_sum * softplus(align_sum/64 + 0.5) / 64
// ---------------------------------------------------------------------------
__global__ __launch_bounds__(256)
void phasor_align_kernel(const float* __restrict__ kraw,
                         const float* __restrict__ qraw,
                         float* __restrict__ scale)
{
    int row  = blockIdx.x * 8 + (threadIdx.x >> 5);
    int lane = threadIdx.x & 31;
    const float* kr = kraw + (size_t)row * PP;
    const float* qr = qraw + (size_t)row * PP;

    float s = 0.0f;
    #pragma unroll
    for (int i = lane; i < PP; i += 32) {
        float kp = tanhf(kr[i]) * PI_F;
        float qp = tanhf(qr[i]) * PI_F;
        s += cosf(kp - qp);
    }
    #pragma unroll
    for (int off = 16; off > 0; off >>= 1)
        s += __shfl_down(s, off, 32);

    if (lane == 0) {
        float z  = s / (float)PP + 0.5f;
        float sp = (z > 20.0f) ? z : log1pf(expf(z));   // softplus
        scale[row] = s * sp / (float)PP;
    }
}

// ---------------------------------------------------------------------------
// In-place LayerNorm over DIM=4096, one 256-thread block per row.
// ---------------------------------------------------------------------------
__global__ __launch_bounds__(256)
void phasor_ln_kernel(float* __restrict__ o,
                      const float* __restrict__ g,
                      const float* __restrict__ b)
{
    float* p = o + (size_t)blockIdx.x * KDIM;
    const int tid = threadIdx.x;

    float vals[16];
    float lsum = 0.0f, lsq = 0.0f;
    #pragma unroll
    for (int i = 0; i < 16; ++i) {
        float v = p[tid + i * 256];
        vals[i] = v;
        lsum += v;
        lsq  += v * v;
    }
    #pragma unroll
    for (int off = 16; off > 0; off >>= 1) {
        lsum += __shfl_down(lsum, off, 32);
        lsq  += __shfl_down(lsq,  off, 32);
    }
    __shared__ float ws1[8], ws2[8];
    int wv = tid >> 5, ln = tid & 31;
    if (ln == 0) { ws1[wv] = lsum; ws2[wv] = lsq; }
    __syncthreads();
    if (tid == 0) {
        float a = 0.0f, c = 0.0f;
        #pragma unroll
        for (int i = 0; i < 8; ++i) { a += ws1[i]; c += ws2[i]; }
        ws1[0] = a; ws2[0] = c;
    }
    __syncthreads();
    float mean = ws1[0] / (float)KDIM;
    float var  = ws2[0] / (float)KDIM - mean * mean;
    float inv  = rsqrtf(var + LN_EPS_F);
    #pragma unroll
    for (int i = 0; i < 16; ++i) {
        int c = tid + i * 256;
        p[c] = (vals[i] - mean) * inv * g[c] + b[c];
    }
}

// ---------------------------------------------------------------------------
extern "C" void kernel_launch(void* const* d_in, const int* in_sizes, int n_in,
                              void* d_out, int out_size, void* d_ws, size_t ws_size,
                              hipStream_t stream)
{
    const float* x   = (const float*)d_in[0];
    const float* Wk  = (const float*)d_in[1];
    const float* bk  = (const float*)d_in[2];
    const float* Wq  = (const float*)d_in[3];
    const float* bq  = (const float*)d_in[4];
    const float* Wv  = (const float*)d_in[5];
    const float* bv  = (const float*)d_in[6];
    const float* lng = (const float*)d_in[7];
    const float* lnb = (const float*)d_in[8];
    const float* Wo  = (const float*)d_in[9];
    const float* bo  = (const float*)d_in[10];
    float* out = (float*)d_out;

    char* ws = (char*)d_ws;
    float* o_buf = (float*)ws;                                    // [B, DIM]
    float* kraw  = (float*)(ws + (size_t)BROWS * KDIM * 4);       // [B, P]
    float* qraw  = kraw + (size_t)BROWS * PP;                     // [B, P]
    float* scale = qraw + (size_t)BROWS * PP;                     // [B]

    dim3 blk(256);

    // 1/2: phase GEMMs (N=64 -> one N-block, 32 M-blocks)
    phasor_gemm_xWT<<<dim3(1, BROWS / 256), blk, 0, stream>>>(
        x, Wk, bk, nullptr, nullptr, kraw, PP);
    phasor_gemm_xWT<<<dim3(1, BROWS / 256), blk, 0, stream>>>(
        x, Wq, bq, nullptr, nullptr, qraw, PP);

    // 3: align_sum -> combined row scale
    phasor_align_kernel<<<BROWS / 8, blk, 0, stream>>>(kraw, qraw, scale);

    // 4: V GEMM with fused (bias, row-scale) epilogue -> pre-LN activations
    phasor_gemm_xWT<<<dim3(KDIM / 64, BROWS / 256), blk, 0, stream>>>(
        x, Wv, bv, scale, nullptr, o_buf, KDIM);

    // 5: in-place LayerNorm
    phasor_ln_kernel<<<BROWS, blk, 0, stream>>>(o_buf, lng, lnb);

    // 6: output GEMM with fused bias + residual(x)
    phasor_gemm_xWT<<<dim3(KDIM / 64, BROWS / 256), blk, 0, stream>>>(
        o_buf, Wo, bo, nullptr, x, out, KDIM);
}